// MeshCNNLayer_46256797778342
// MI455X (gfx1250) — compile-verified
//
#include <hip/hip_runtime.h>

// ---------------------------------------------------------------------------
// MeshCNN-like GAT layer for MI455X (gfx1250, wave32).
//
// Key rewrite: scores = ((xi+xj) @ W^T) @ a  ==  (xi+xj) . (W^T a)
//   => v = a @ W  (computed on the WMMA pipe, fp32 16x16x4)
//   => p[n] = x[n] . v  (per-node scalar)
//   => score_e = leaky_relu(p[src]+p[dst])
// Removes the [E,256]x[256,256] GEMM (105 GFLOP -> ~0.6 GFLOP); the kernel
// becomes L2-bandwidth bound (x = 51.2MB lives in the 192MB L2).
// ---------------------------------------------------------------------------

typedef __attribute__((ext_vector_type(2))) float v2f;
typedef __attribute__((ext_vector_type(8))) float v8f;

#define DIM 256
#define RB  256   // reduction grid blocks

// ---------------------------------------------------------------------------
// K0: v = a @ W   (v[j] = sum_k a[k] * W[k*DIM + j]) on the matrix pipe.
// One block of 512 threads = 16 waves; wave w owns output columns [16w,16w+16).
// A tile (16x4 f32, ISA layout): VGPR0 holds K=0 (lanes 0-15) / K=2 (16-31),
// VGPR1 holds K=1 / K=3. Only row M=0 is nonzero (= chunk of a).
// B tile (4x16) mirrored: VGPR0 = rows K=0/K=2, VGPR1 = rows K=1/K=3.
// D row M=0 lands in acc[0], lanes 0-15.
// ---------------------------------------------------------------------------
__global__ void k_vec_wmma(const float* __restrict__ W,
                           const float* __restrict__ a,
                           float* __restrict__ v) {
  const int wave = threadIdx.x >> 5;   // 0..15 -> N tile
  const int lane = threadIdx.x & 31;
  const int col  = lane & 15;
  const int hi   = lane >> 4;          // 0: K pair {0,1}, 1: {2,3}
  const int ncol = wave * 16 + col;
  v8f acc = {};
  for (int k0 = 0; k0 < DIM; k0 += 4) {
    v2f A = {};
    if (lane == 0)  { A.x = a[k0 + 0]; A.y = a[k0 + 1]; }
    if (lane == 16) { A.x = a[k0 + 2]; A.y = a[k0 + 3]; }
    v2f B;
    B.x = W[(k0 + 2 * hi + 0) * DIM + ncol];
    B.y = W[(k0 + 2 * hi + 1) * DIM + ncol];
    acc = __builtin_amdgcn_wmma_f32_16x16x4_f32(false, A, false, B,
                                                (short)0, acc, false, false);
  }
  if (lane < 16) v[ncol] = acc[0];
}

// K1: p[n] = x[n] . v  — one wave per node, float4 coalesced loads.
__global__ void k_node_p(const float* __restrict__ x, const float* __restrict__ v,
                         float* __restrict__ p, int N) {
  const int wave = threadIdx.x >> 5;
  const int lane = threadIdx.x & 31;
  const int n = blockIdx.x * 8 + wave;
  if (n >= N) return;
  const float4* xr = (const float4*)(x + (size_t)n * DIM);
  const float4* vr = (const float4*)v;
  float acc = 0.f;
#pragma unroll
  for (int i = 0; i < 2; ++i) {
    float4 xv = xr[i * 32 + lane];
    float4 vv = vr[i * 32 + lane];
    acc += xv.x * vv.x + xv.y * vv.y + xv.z * vv.z + xv.w * vv.w;
  }
#pragma unroll
  for (int off = 16; off > 0; off >>= 1) acc += __shfl_xor(acc, off, 32);
  if (lane == 0) p[n] = acc;
}

// K2: per edge, one wave: s[e] = leaky(p[src]+p[dst]), ef[e] = ||xi - xj||.
__global__ void k_edge_score(const float* __restrict__ x, const int* __restrict__ ei,
                             const float* __restrict__ p,
                             float* __restrict__ s, float* __restrict__ ef, int E) {
  const int wave = threadIdx.x >> 5;
  const int lane = threadIdx.x & 31;
  const int e = blockIdx.x * 8 + wave;
  if (e >= E) return;
  const int src = ei[e], dst = ei[E + e];
  const float4* xi = (const float4*)(x + (size_t)src * DIM);
  const float4* xj = (const float4*)(x + (size_t)dst * DIM);
  float acc = 0.f;
#pragma unroll
  for (int i = 0; i < 2; ++i) {
    float4 av = xi[i * 32 + lane], bv = xj[i * 32 + lane];
    float dx = av.x - bv.x, dy = av.y - bv.y, dz = av.z - bv.z, dw = av.w - bv.w;
    acc += dx * dx + dy * dy + dz * dz + dw * dw;
  }
#pragma unroll
  for (int off = 16; off > 0; off >>= 1) acc += __shfl_xor(acc, off, 32);
  if (lane == 0) {
    float sr = p[src] + p[dst];
    s[e]  = sr > 0.f ? sr : 0.2f * sr;   // leaky_relu, slope 0.2
    ef[e] = __builtin_sqrtf(acc);
  }
}

__global__ void k_max_part(const float* __restrict__ s, int E, float* __restrict__ part) {
  __shared__ float sm[256];
  float m = -__builtin_inff();
  for (int i = blockIdx.x * 256 + threadIdx.x; i < E; i += RB * 256)
    m = fmaxf(m, s[i]);
  sm[threadIdx.x] = m; __syncthreads();
  for (int w = 128; w > 0; w >>= 1) {
    if (threadIdx.x < w) sm[threadIdx.x] = fmaxf(sm[threadIdx.x], sm[threadIdx.x + w]);
    __syncthreads();
  }
  if (threadIdx.x == 0) part[blockIdx.x] = sm[0];
}

__global__ void k_max_final(const float* __restrict__ part, float* __restrict__ mout) {
  __shared__ float sm[256];
  sm[threadIdx.x] = part[threadIdx.x]; __syncthreads();
  for (int w = 128; w > 0; w >>= 1) {
    if (threadIdx.x < w) sm[threadIdx.x] = fmaxf(sm[threadIdx.x], sm[threadIdx.x + w]);
    __syncthreads();
  }
  if (threadIdx.x == 0) mout[0] = sm[0];
}

__global__ void k_sum_part(const float* __restrict__ s, int E,
                           const float* __restrict__ m, float* __restrict__ part) {
  __shared__ float sm[256];
  const float mm = m[0];
  float acc = 0.f;
  for (int i = blockIdx.x * 256 + threadIdx.x; i < E; i += RB * 256)
    acc += __expf(s[i] - mm);
  sm[threadIdx.x] = acc; __syncthreads();
  for (int w = 128; w > 0; w >>= 1) {
    if (threadIdx.x < w) sm[threadIdx.x] += sm[threadIdx.x + w];
    __syncthreads();
  }
  if (threadIdx.x == 0) part[blockIdx.x] = sm[0];
}

__global__ void k_sum_final(const float* __restrict__ part, float* __restrict__ invZ) {
  __shared__ float sm[256];
  sm[threadIdx.x] = part[threadIdx.x]; __syncthreads();
  for (int w = 128; w > 0; w >>= 1) {
    if (threadIdx.x < w) sm[threadIdx.x] += sm[threadIdx.x + w];
    __syncthreads();
  }
  if (threadIdx.x == 0) invZ[0] = 1.f / sm[0];
}

// Fold softmax weight * edge_feature into ef[] (one expf per edge).
__global__ void k_wgt(const float* __restrict__ s, float* __restrict__ ef,
                      const float* __restrict__ m, const float* __restrict__ invZ, int E) {
  const int e = blockIdx.x * 256 + threadIdx.x;
  if (e < E) ef[e] = __expf(s[e] - m[0]) * invZ[0] * ef[e];
}

// K7: out[src] += w_e * x[dst]  — 8 edges/block, 256 lanes = 256 columns.
__global__ void k_scatter(const float* __restrict__ x, const int* __restrict__ ei,
                          const float* __restrict__ wgt, float* __restrict__ out, int E) {
  __shared__ float wsh[8];
  __shared__ int   ssh[8], dsh[8];
  const int e0 = blockIdx.x * 8;
  const int tid = threadIdx.x;
  if (tid < 8) {
    const int e = e0 + tid;
    if (e < E) { wsh[tid] = wgt[e]; ssh[tid] = ei[e]; dsh[tid] = ei[E + e]; }
    else       { wsh[tid] = 0.f;    ssh[tid] = 0;     dsh[tid] = 0; }
  }
  __syncthreads();
#pragma unroll
  for (int k = 0; k < 8; ++k) {
    const int e = e0 + k;
    if (e >= E) break;
    const float val = wsh[k] * x[(size_t)dsh[k] * DIM + tid];
    // relaxed agent-scope fp32 add -> global_atomic_add_f32
    __hip_atomic_fetch_add(&out[(size_t)ssh[k] * DIM + tid], val,
                           __ATOMIC_RELAXED, __HIP_MEMORY_SCOPE_AGENT);
  }
}

__global__ void k_zero(float* __restrict__ out, size_t n) {
  const size_t i = (size_t)blockIdx.x * blockDim.x + threadIdx.x;
  if (i < n) out[i] = 0.f;
}

extern "C" void kernel_launch(void* const* d_in, const int* in_sizes, int n_in,
                              void* d_out, int out_size, void* d_ws, size_t ws_size,
                              hipStream_t stream) {
  const float* x  = (const float*)d_in[0];   // [N, 256]
  const int*   ei = (const int*)  d_in[1];   // [2, E] (row0 = src, row1 = dst)
  const float* W  = (const float*)d_in[2];   // [256, 256]
  const float* a  = (const float*)d_in[3];   // [256]
  float* out = (float*)d_out;                // [N, 256]

  const int N = in_sizes[0] / DIM;
  const int E = in_sizes[1] / 2;

  // Workspace layout (floats): ~6.6 MB total
  float* v    = (float*)d_ws;     // DIM
  float* p    = v + DIM;          // N
  float* s    = p + N;            // E  (scores, post-leaky)
  float* ef   = s + E;            // E  (edge norm, later final weight)
  float* part = ef + E;           // RB
  float* mval = part + RB;        // 1
  float* invZ = mval + 1;         // 1

  k_zero     <<<(unsigned)(((size_t)N * DIM + 255) / 256), 256, 0, stream>>>(out, (size_t)N * DIM);
  k_vec_wmma <<<1, 512, 0, stream>>>(W, a, v);
  k_node_p   <<<(N + 7) / 8, 256, 0, stream>>>(x, v, p, N);
  k_edge_score<<<(E + 7) / 8, 256, 0, stream>>>(x, ei, p, s, ef, E);
  k_max_part <<<RB, 256, 0, stream>>>(s, E, part);
  k_max_final<<<1, 256, 0, stream>>>(part, mval);
  k_sum_part <<<RB, 256, 0, stream>>>(s, E, mval, part);
  k_sum_final<<<1, 256, 0, stream>>>(part, invZ);
  k_wgt      <<<(E + 255) / 256, 256, 0, stream>>>(s, ef, mval, invZ, E);
  k_scatter  <<<(E + 7) / 8, 256, 0, stream>>>(x, ei, ef, out, E);
}